// GNNModel_7258494730681
// MI455X (gfx1250) — compile-verified
//
#include <hip/hip_runtime.h>
#include <math.h>

// ---------------- problem constants (match reference) ----------------
#define NNODES 50000
#define NEDGES 1600000
#define F_IN   13
#define KPAD   16
#define HID    64
#define ED     15
#define NGRAPH 512
#define NEG_SLOPE 0.2f

typedef __attribute__((ext_vector_type(2))) float v2f;
typedef __attribute__((ext_vector_type(8))) float v8f;

// order-preserving float <-> uint for atomic max
__device__ __forceinline__ unsigned fenc(float f) {
    unsigned u = __float_as_uint(f);
    return (u & 0x80000000u) ? ~u : (u | 0x80000000u);
}
__device__ __forceinline__ float fdec(unsigned u) {
    return __uint_as_float((u & 0x80000000u) ? (u & 0x7FFFFFFFu) : ~u);
}

// ---------------- zero-pad x:[N,13]->[N,16] and W1:[13,64]->[16,64] -------------
__global__ void gat_pad_x(const float* __restrict__ x, float* __restrict__ xpad) {
    int t = blockIdx.x * blockDim.x + threadIdx.x;
    if (t >= NNODES * KPAD) return;
    int n = t >> 4, j = t & 15;
    xpad[t] = (j < F_IN) ? x[(size_t)n * F_IN + j] : 0.0f;
}
__global__ void gat_pad_w(const float* __restrict__ W, float* __restrict__ Wpad) {
    int t = blockIdx.x * blockDim.x + threadIdx.x;
    if (t >= KPAD * HID) return;
    Wpad[t] = (t < F_IN * HID) ? W[t] : 0.0f;   // row-major: leading rows coincide
}

// ---------------- self-loop attr = mean of incoming edge attrs ----------------
__global__ void gat_loop_accum(const int* __restrict__ dst,
                               const float* __restrict__ edge_attr,
                               float* __restrict__ loop_sum,
                               float* __restrict__ cnt) {
    long long tid = (long long)blockIdx.x * blockDim.x + threadIdx.x;
    long long e = tid >> 4;
    int j = (int)(tid & 15);
    if (e >= NEDGES) return;
    int d = dst[e];
    if (j < ED) atomicAdd(&loop_sum[(size_t)d * ED + j], edge_attr[(size_t)e * ED + j]);
    else if (j == ED) atomicAdd(&cnt[d], 1.0f);
}

__global__ void gat_loop_div(float* __restrict__ loop_sum, const float* __restrict__ cnt) {
    int t = blockIdx.x * blockDim.x + threadIdx.x;
    if (t >= NNODES * ED) return;
    int n = t / ED;
    loop_sum[t] /= fmaxf(cnt[n], 1.0f);
}

// ---------------- fp32 WMMA GEMM: C[M x 64] = A[M x K] @ B[K x 64] ----------------
// K is compile-time (16 or 64) and a multiple of 4 -> fully unrolled, guard-free.
// One wave -> 16x16 tile via V_WMMA_F32_16X16X4_F32; 4 waves cover HID=64 cols.
// B staged in LDS once per block.
template <int K>
__global__ __launch_bounds__(128) void gat_gemm_wmma(const float* __restrict__ A,
                                                     const float* __restrict__ B,
                                                     float* __restrict__ C) {
    __shared__ float Bs[K * HID];
    constexpr int TOT4 = K * HID / 4;
    for (int i = threadIdx.x; i < TOT4; i += 128)
        ((float4*)Bs)[i] = ((const float4*)B)[i];
    __syncthreads();

    const int wave = threadIdx.x >> 5;       // 0..3 -> column tile
    const int lane = threadIdx.x & 31;
    const int row0 = blockIdx.x * 16;        // M = 50000 = 3125*16, exact
    const int col0 = wave * 16;
    const int r    = lane & 15;
    const int half = lane >> 4;              // 0 or 1
    const int cb   = col0 + r;

    v8f acc = {};
    const float* Arow = A + (size_t)(row0 + r) * K + half * 2;
#pragma unroll
    for (int k = 0; k < K; k += 4) {
        v2f a = *(const v2f*)(Arow + k);                 // aligned b64 load
        int kb = k + half * 2;
        v2f b;
        b.x = Bs[kb * HID + cb];
        b.y = Bs[(kb + 1) * HID + cb];
        acc = __builtin_amdgcn_wmma_f32_16x16x4_f32(false, a, false, b,
                                                    (short)0, acc, false, false);
    }
    float* Cp = C + (size_t)row0 * HID + col0;
#pragma unroll
    for (int i = 0; i < 8; ++i) {
        int rr = i + half * 8;               // C/D layout: vgpr i holds M=i / M=i+8
        Cp[(size_t)rr * HID + r] = acc[i];
    }
}

// ---------------- per-node attention logits alpha_s / alpha_d ----------------
__global__ void gat_alpha_sd(const float* __restrict__ h,
                             const float* __restrict__ a_src,
                             const float* __restrict__ a_dst,
                             float* __restrict__ as, float* __restrict__ ad) {
    int n = blockIdx.x * blockDim.x + threadIdx.x;
    if (n >= NNODES) return;
    const float* hp = h + (size_t)n * HID;
    float s = 0.f, d = 0.f;
#pragma unroll
    for (int c = 0; c < HID; ++c) { float v = hp[c]; s += v * a_src[c]; d += v * a_dst[c]; }
    as[n] = s; ad[n] = d;
}

// ---------------- wea = W_e @ a_e  (collapses edge-attr GEMM to mat-vec) --------
__global__ void gat_we_ae(const float* __restrict__ We, const float* __restrict__ ae,
                          float* __restrict__ wea) {
    int i = threadIdx.x;
    if (i >= ED) return;
    float s = 0.f;
#pragma unroll
    for (int c = 0; c < HID; ++c) s += We[(size_t)i * HID + c] * ae[c];
    wea[i] = s;
}

// ---------------- alpha = leaky_relu(as[src]+ad[dst]+ea.wea); segment max -------
__global__ void gat_alpha_max(const int* __restrict__ src, const int* __restrict__ dst,
                              const float* __restrict__ edge_attr,
                              const float* __restrict__ loop_attr,
                              const float* __restrict__ wea,
                              const float* __restrict__ as, const float* __restrict__ ad,
                              float* __restrict__ alpha_arr, unsigned* __restrict__ amax) {
    int t = blockIdx.x * blockDim.x + threadIdx.x;
    if (t >= NEDGES + NNODES) return;
    int s, d; const float* ea;
    if (t < NEDGES) { s = src[t]; d = dst[t]; ea = edge_attr + (size_t)t * ED; }
    else            { s = d = t - NEDGES;     ea = loop_attr + (size_t)(t - NEDGES) * ED; }
    float av = 0.f;
#pragma unroll
    for (int j = 0; j < ED; ++j) av += ea[j] * wea[j];
    float a = as[s] + ad[d] + av;
    a = (a >= 0.f) ? a : NEG_SLOPE * a;
    alpha_arr[t] = a;
    atomicMax(&amax[d], fenc(a));
}

// ---------------- ex = exp(alpha - amax[dst]); denom = segment sum --------------
__global__ void gat_exp_denom(const int* __restrict__ dst,
                              const float* __restrict__ alpha_arr,
                              const unsigned* __restrict__ amax,
                              float* __restrict__ ex_arr, float* __restrict__ denom) {
    int t = blockIdx.x * blockDim.x + threadIdx.x;
    if (t >= NEDGES + NNODES) return;
    int d = (t < NEDGES) ? dst[t] : (t - NEDGES);
    float ex = __expf(alpha_arr[t] - fdec(amax[d]));
    ex_arr[t] = ex;
    atomicAdd(&denom[d], ex);
}

// ---------------- out[dst] += (ex/denom[dst]) * h[src] --------------------------
__global__ void gat_scatter(const int* __restrict__ src, const int* __restrict__ dst,
                            const float* __restrict__ ex_arr, const float* __restrict__ denom,
                            const float* __restrict__ h, float* __restrict__ out) {
    long long tid = (long long)blockIdx.x * blockDim.x + threadIdx.x;
    long long total = (long long)(NEDGES + NNODES) * 16;
    if (tid >= total) return;
    int t = (int)(tid >> 4);
    int c = ((int)tid & 15) * 4;
    int s, d;
    if (t < NEDGES) { s = src[t]; d = dst[t]; }
    else            { s = d = t - NEDGES; }
    float coef = ex_arr[t] / denom[d];
    const float* hs = h + (size_t)s * HID + c;
    float* od = out + (size_t)d * HID + c;
    atomicAdd(od + 0, coef * hs[0]);
    atomicAdd(od + 1, coef * hs[1]);
    atomicAdd(od + 2, coef * hs[2]);
    atomicAdd(od + 3, coef * hs[3]);
}

// ---------------- + bias (and optional relu) ------------------------------------
__global__ void gat_bias_act(float* __restrict__ out, const float* __restrict__ b, int do_relu) {
    int t = blockIdx.x * blockDim.x + threadIdx.x;
    if (t >= NNODES * HID) return;
    float v = out[t] + b[t & (HID - 1)];
    out[t] = do_relu ? fmaxf(v, 0.f) : v;
}

// ---------------- global mean pool ----------------------------------------------
__global__ void gat_pool(const int* __restrict__ batch, const float* __restrict__ h,
                         float* __restrict__ pooled, float* __restrict__ gcnt) {
    int t = blockIdx.x * blockDim.x + threadIdx.x;
    if (t >= NNODES * HID) return;
    int n = t >> 6, c = t & 63;
    int g = batch[n];
    atomicAdd(&pooled[(size_t)g * HID + c], h[t]);
    if (c == 0) atomicAdd(&gcnt[g], 1.0f);
}

__global__ void gat_head(const float* __restrict__ pooled, const float* __restrict__ gcnt,
                         const float* __restrict__ lw, const float* __restrict__ lb,
                         float* __restrict__ out) {
    int g = blockIdx.x * blockDim.x + threadIdx.x;
    if (g >= NGRAPH) return;
    float inv = 1.0f / fmaxf(gcnt[g], 1.0f);
    float acc = lb[0];
#pragma unroll
    for (int c = 0; c < HID; ++c) acc += pooled[(size_t)g * HID + c] * inv * lw[c];
    out[g] = tanhf(acc);
}

// ================================================================================
extern "C" void kernel_launch(void* const* d_in, const int* in_sizes, int n_in,
                              void* d_out, int out_size, void* d_ws, size_t ws_size,
                              hipStream_t stream) {
    const float* x        = (const float*)d_in[0];
    const int*   eidx     = (const int*)d_in[1];     // [2, E]
    const float* eattr    = (const float*)d_in[2];
    const int*   batch    = (const int*)d_in[3];
    const float* W1       = (const float*)d_in[4];
    const float* asrc1    = (const float*)d_in[5];
    const float* adst1    = (const float*)d_in[6];
    const float* We1      = (const float*)d_in[7];
    const float* ae1      = (const float*)d_in[8];
    const float* b1       = (const float*)d_in[9];
    const float* W2       = (const float*)d_in[10];
    const float* asrc2    = (const float*)d_in[11];
    const float* adst2    = (const float*)d_in[12];
    const float* We2      = (const float*)d_in[13];
    const float* ae2      = (const float*)d_in[14];
    const float* b2       = (const float*)d_in[15];
    const float* lin_w    = (const float*)d_in[16];
    const float* lin_b    = (const float*)d_in[17];
    float* out = (float*)d_out;
    const int* src = eidx;
    const int* dst = eidx + NEDGES;

    // -------- workspace carve-up (256B aligned) --------
    char* w = (char*)d_ws;
    size_t off = 0;
    auto carve = [&](size_t bytes) -> char* {
        char* p = w + off;
        off += (bytes + 255) & ~(size_t)255;
        return p;
    };
    float*    loop_attr = (float*)carve((size_t)NNODES * ED * 4);
    float*    cnt       = (float*)carve((size_t)NNODES * 4);
    float*    xpad      = (float*)carve((size_t)NNODES * KPAD * 4);
    float*    W1pad     = (float*)carve((size_t)KPAD * HID * 4);
    float*    hbuf      = (float*)carve((size_t)NNODES * HID * 4);   // projections
    float*    xbuf      = (float*)carve((size_t)NNODES * HID * 4);   // conv1 out / layer2 in
    float*    obuf      = (float*)carve((size_t)NNODES * HID * 4);   // conv2 out
    float*    asbuf     = (float*)carve((size_t)NNODES * 4);
    float*    adbuf     = (float*)carve((size_t)NNODES * 4);
    unsigned* amax      = (unsigned*)carve((size_t)NNODES * 4);
    float*    denom     = (float*)carve((size_t)NNODES * 4);
    float*    alpha_arr = (float*)carve((size_t)(NEDGES + NNODES) * 4);
    float*    ex_arr    = (float*)carve((size_t)(NEDGES + NNODES) * 4);
    float*    wea       = (float*)carve(64 * 4);
    float*    pooled    = (float*)carve((size_t)NGRAPH * HID * 4);
    float*    gcnt      = (float*)carve((size_t)NGRAPH * 4);
    (void)ws_size; (void)in_sizes; (void)n_in; (void)out_size;

    const int TB = 256;
    const long long teTot = (long long)(NEDGES + NNODES);
    const int gridEN   = (int)((teTot + TB - 1) / TB);
    const int gridEN16 = (int)((teTot * 16 + TB - 1) / TB);
    const int gridE16  = (int)(((long long)NEDGES * 16 + TB - 1) / TB);
    const int gridNH   = (NNODES * HID + TB - 1) / TB;
    const int gridN    = (NNODES + TB - 1) / TB;
    const int gridM    = NNODES / 16;   // 3125, exact

    // -------- pad x / W1 for guard-free WMMA (K=16) --------
    gat_pad_x<<<(NNODES * KPAD + TB - 1) / TB, TB, 0, stream>>>(x, xpad);
    gat_pad_w<<<(KPAD * HID + TB - 1) / TB, TB, 0, stream>>>(W1, W1pad);

    // -------- self-loop edge attrs (shared by both layers) --------
    hipMemsetAsync(loop_attr, 0, (size_t)NNODES * ED * 4, stream);
    hipMemsetAsync(cnt, 0, (size_t)NNODES * 4, stream);
    gat_loop_accum<<<gridE16, TB, 0, stream>>>(dst, eattr, loop_attr, cnt);
    gat_loop_div<<<(NNODES * ED + TB - 1) / TB, TB, 0, stream>>>(loop_attr, cnt);

    // ================= layer 1 =================
    gat_gemm_wmma<KPAD><<<gridM, 128, 0, stream>>>(xpad, W1pad, hbuf);
    gat_alpha_sd<<<gridN, TB, 0, stream>>>(hbuf, asrc1, adst1, asbuf, adbuf);
    gat_we_ae<<<1, 64, 0, stream>>>(We1, ae1, wea);
    hipMemsetAsync(amax, 0, (size_t)NNODES * 4, stream);
    hipMemsetAsync(denom, 0, (size_t)NNODES * 4, stream);
    hipMemsetAsync(xbuf, 0, (size_t)NNODES * HID * 4, stream);
    gat_alpha_max<<<gridEN, TB, 0, stream>>>(src, dst, eattr, loop_attr, wea,
                                             asbuf, adbuf, alpha_arr, amax);
    gat_exp_denom<<<gridEN, TB, 0, stream>>>(dst, alpha_arr, amax, ex_arr, denom);
    gat_scatter<<<gridEN16, TB, 0, stream>>>(src, dst, ex_arr, denom, hbuf, xbuf);
    gat_bias_act<<<gridNH, TB, 0, stream>>>(xbuf, b1, 1);

    // ================= layer 2 =================
    gat_gemm_wmma<HID><<<gridM, 128, 0, stream>>>(xbuf, W2, hbuf);
    gat_alpha_sd<<<gridN, TB, 0, stream>>>(hbuf, asrc2, adst2, asbuf, adbuf);
    gat_we_ae<<<1, 64, 0, stream>>>(We2, ae2, wea);
    hipMemsetAsync(amax, 0, (size_t)NNODES * 4, stream);
    hipMemsetAsync(denom, 0, (size_t)NNODES * 4, stream);
    hipMemsetAsync(obuf, 0, (size_t)NNODES * HID * 4, stream);
    gat_alpha_max<<<gridEN, TB, 0, stream>>>(src, dst, eattr, loop_attr, wea,
                                             asbuf, adbuf, alpha_arr, amax);
    gat_exp_denom<<<gridEN, TB, 0, stream>>>(dst, alpha_arr, amax, ex_arr, denom);
    gat_scatter<<<gridEN16, TB, 0, stream>>>(src, dst, ex_arr, denom, hbuf, obuf);
    gat_bias_act<<<gridNH, TB, 0, stream>>>(obuf, b2, 0);

    // ================= pool + head =================
    hipMemsetAsync(pooled, 0, (size_t)NGRAPH * HID * 4, stream);
    hipMemsetAsync(gcnt, 0, (size_t)NGRAPH * 4, stream);
    gat_pool<<<gridNH, TB, 0, stream>>>(batch, obuf, pooled, gcnt);
    gat_head<<<(NGRAPH + 63) / 64, 64, 0, stream>>>(pooled, gcnt, lin_w, lin_b, out);
}